// AgentCollisionLoss_50105088475723
// MI455X (gfx1250) — compile-verified
//
#include <hip/hip_runtime.h>
#include <math.h>

typedef float v2f __attribute__((ext_vector_type(2)));
typedef float v8f __attribute__((ext_vector_type(8)));

#define KPAD  8
#define NAG   64
#define TT    80
#define BBATCH 4
#define ROWS  (NAG * KPAD)   // 512 padded disk rows per (b,t)
#define TILES (ROWS / 16)    // 32 tiles per dimension

// DPP8 selector: arbitrary permutation within each group of 8 lanes.
#define DPP8SEL(a,b,c,d,e,f,g,h) \
  ((a)|((b)<<3)|((c)<<6)|((d)<<9)|((e)<<12)|((f)<<15)|((g)<<18)|((h)<<21))

// Min over each aligned group of 8 lanes — pure VALU (v_mov_b32_dpp8), no waits.
__device__ __forceinline__ float min8_group(float x)
{
#if __has_builtin(__builtin_amdgcn_mov_dpp8)
    float y;
    y = __int_as_float(__builtin_amdgcn_mov_dpp8(__float_as_int(x),
            DPP8SEL(1,0,3,2,5,4,7,6)));                  // xor 1
    x = fminf(x, y);
    y = __int_as_float(__builtin_amdgcn_mov_dpp8(__float_as_int(x),
            DPP8SEL(2,3,0,1,6,7,4,5)));                  // xor 2
    x = fminf(x, y);
    y = __int_as_float(__builtin_amdgcn_mov_dpp8(__float_as_int(x),
            DPP8SEL(4,5,6,7,0,1,2,3)));                  // xor 4
    x = fminf(x, y);
#else
    x = fminf(x, __shfl_xor(x, 1, 32));
    x = fminf(x, __shfl_xor(x, 2, 32));
    x = fminf(x, __shfl_xor(x, 4, 32));
#endif
    return x;
}

__device__ __forceinline__ float fast_sqrt(float x)
{
#if __has_builtin(__builtin_amdgcn_sqrtf)
    return __builtin_amdgcn_sqrtf(x);   // raw v_sqrt_f32
#else
    return sqrtf(x);
#endif
}

__device__ __forceinline__ float fast_rcp(float x)
{
#if __has_builtin(__builtin_amdgcn_rcpf)
    return __builtin_amdgcn_rcpf(x);    // raw v_rcp_f32
#else
    return 1.0f / x;
#endif
}

// Per-tile epilogue: min over the 8x8 disk block, penalty for the agent pair.
__device__ __forceinline__ float tile_pen(const v8f& c, float pd,
                                          bool offdiag, bool leader)
{
    // In-lane min over this lane's 8 rows (raw, clamp once afterwards:
    // min(max(x,0)) == max(min(x),0)).
    float m = c[0];
    #pragma unroll
    for (int q = 1; q < 8; ++q) m = fminf(m, c[q]);
    m = min8_group(m);                  // min over the 8 columns of the group
    float mind2 = fmaxf(m, 0.0f);
    float mind  = fast_sqrt(mind2);
    mind = (mind2 > 1e-12f) ? mind : 0.0f;
    float pen = 1.0f - mind * fast_rcp(pd);
    bool  ok  = offdiag && (mind <= pd) && leader;
    return ok ? pen : 0.0f;
}

// One workgroup per (b,t). 256 threads = 8 waves (wave32).
__global__ void __launch_bounds__(256)
collision_loss_kernel(const float* __restrict__ Y,
                      const float* __restrict__ length,
                      const float* __restrict__ width,
                      float* __restrict__ out)
{
    __shared__ float lA[ROWS][4];              // [x, y, |w|^2, 1]
    __shared__ float lB[ROWS][4];              // [-2x, -2y, 1, |w|^2]
    __shared__ __align__(16) float lrad[NAG];
    __shared__ float wsum[8];

    const int bt   = blockIdx.x;
    const int b    = bt / TT;
    const int t    = bt % TT;
    const int tid  = threadIdx.x;
    const int lane = tid & 31;
    const int wave = tid >> 5;

    if (tid < NAG) lrad[tid] = 0.5f * width[b * NAG + tid];

    // Augmented rows: disk k of agent i -> row i*8+k (k=5..7 duplicate k=4;
    // duplicated points cannot change a pairwise min distance).
    for (int r = tid; r < ROWS; r += 256) {
        int i = r >> 3;
        int k = r & 7; if (k > 4) k = 4;
        const float* y = Y + (((size_t)(b * NAG + i) * TT + t) * 6);
        float px  = y[0], py = y[1], yaw = y[4];
        float len = length[b * NAG + i];
        float rad = 0.5f * width[b * NAG + i];
        float cmin = -0.5f * len + rad;
        float cmax =  0.5f * len - rad;
        float cx   = cmin + (cmax - cmin) * (0.25f * (float)k);
        float cs = cosf(yaw), sn = sinf(yaw);
        float wx =  cx * cs + px;
        float wy = -cx * sn + py;
        float sq = wx * wx + wy * wy;
        lA[r][0] = wx;          lA[r][1] = wy;
        lA[r][2] = sq;          lA[r][3] = 1.0f;
        lB[r][0] = -2.0f * wx;  lB[r][1] = -2.0f * wy;
        lB[r][2] = 1.0f;        lB[r][3] = sq;
    }
    __syncthreads();

    // WMMA f32 16x16x4 fragment addressing (wave32):
    //  A: lane<16 -> M=lane, K=0..1 ; lane>=16 -> M=lane-16, K=2..3 (same for B/N).
    const int  halfsel = lane >> 4;
    const int  rc      = lane & 15;
    const int  koff    = halfsel * 2;
    const int  colhalf = (lane >> 3) & 1;      // which col-agent of the tile
    const bool leader  = (lane & 7) == 0;      // one lane per 8-lane group

    float acc = 0.0f;

    for (int ti = wave; ti < TILES; ti += 8) {
        v2f a;
        a.x = lA[ti * 16 + rc][koff + 0];
        a.y = lA[ti * 16 + rc][koff + 1];
        const int   a_agent = 2 * ti + halfsel;
        const float rad_a02 = lrad[a_agent] + 0.2f;

        for (int tj = 0; tj < TILES; tj += 2) {
            v2f b0, b1;
            b0.x = lB[(tj + 0) * 16 + rc][koff + 0];
            b0.y = lB[(tj + 0) * 16 + rc][koff + 1];
            b1.x = lB[(tj + 1) * 16 + rc][koff + 0];
            b1.y = lB[(tj + 1) * 16 + rc][koff + 1];

            // 4 radii for the two column tiles in one aligned b128 load.
            float4 rb = *(const float4*)&lrad[2 * tj];
            float  pd0 = rad_a02 + (colhalf ? rb.y : rb.x);
            float  pd1 = rad_a02 + (colhalf ? rb.w : rb.z);

            v8f c0 = {};
            v8f c1 = {};
            // D(m,n) = [x,y,sq,1]_m . [-2x,-2y,1,sq]_n = squared distance
            c0 = __builtin_amdgcn_wmma_f32_16x16x4_f32(
                     false, a, false, b0, (short)0, c0, false, false);
            c1 = __builtin_amdgcn_wmma_f32_16x16x4_f32(
                     false, a, false, b1, (short)0, c1, false, false);

            acc += tile_pen(c0, pd0, a_agent != 2 * tj + 0 * 2 + colhalf, leader);
            acc += tile_pen(c1, pd1, a_agent != 2 * (tj + 1) + colhalf,   leader);
        }
    }

    // Sum across the wave, then across waves, one global atomic per block.
    for (int m = 16; m >= 1; m >>= 1) acc += __shfl_xor(acc, m, 32);
    if (lane == 0) wsum[wave] = acc;
    __syncthreads();

    if (tid == 0) {
        float total = 0.0f;
        for (int wv = 0; wv < 8; ++wv) total += wsum[wv];
        float norm  = (1.0f - powf(0.9f, (float)TT)) / (1.0f - 0.9f);
        float scale = powf(0.9f, (float)t) / norm / (float)(BBATCH * NAG * TT);
        atomicAdd(out, total * scale);
    }
}

__global__ void zero_out_kernel(float* out)
{
    if (threadIdx.x == 0) out[0] = 0.0f;
}

extern "C" void kernel_launch(void* const* d_in, const int* in_sizes, int n_in,
                              void* d_out, int out_size, void* d_ws, size_t ws_size,
                              hipStream_t stream)
{
    const float* Y      = (const float*)d_in[0];
    const float* length = (const float*)d_in[1];
    const float* width  = (const float*)d_in[2];
    float*       out    = (float*)d_out;

    zero_out_kernel<<<dim3(1), dim3(32), 0, stream>>>(out);
    collision_loss_kernel<<<dim3(BBATCH * TT), dim3(256), 0, stream>>>(
        Y, length, width, out);
}